// Thalamus_90314572300858
// MI455X (gfx1250) — compile-verified
//
#include <hip/hip_runtime.h>
#include <cstddef>

typedef float v2f __attribute__((ext_vector_type(2)));
typedef float v4f __attribute__((ext_vector_type(4)));
typedef float v8f __attribute__((ext_vector_type(8)));

#define Bsz 16
#define Ssz 1024
#define Dsz 1024
#define Hsz 256
#define Esz 8
#define Ktop 3
#define Msz (Bsz * Ssz)   // 16384 tokens

#define APAD 34           // A row stride in LDS (even -> 8B-aligned k-pairs, conflict-free)
#define BPAD 288          // B kpair-row stride (256 data + 32 pad -> halves hit disjoint banks)

// ---------------------------------------------------------------------------
// Kernel 1: gate GEMM  Y = X @ Wg  fused with  gated = x * sigmoid(Y + bg)
// Block: 256 threads = 8 waves. Block tile 64(M) x 128(N).
// Waves 4(M) x 2(N); each wave owns 16x64 -> 4 v8f accumulators.
// Double-buffered LDS, K chunks of 32, fp32 WMMA (16x16x4).
// A in LDS: [row][k] stride 34  -> A frag = 1 aligned ds_load_b64.
// B in LDS: [kpair][n*2 + k&1]  -> B frag = 1 aligned ds_load_b64.
// ---------------------------------------------------------------------------
__global__ __launch_bounds__(256) void gate_gemm_kernel(
    const float* __restrict__ x, const float* __restrict__ Wg,
    const float* __restrict__ bg, float* __restrict__ gated)
{
    __shared__ __align__(16) float A_s[2][64 * APAD];
    __shared__ __align__(16) float B_s[2][16 * BPAD];

    const int tid  = threadIdx.x;
    const int lane = tid & 31;
    const int wave = tid >> 5;
    const int half = lane >> 4;         // 0 = lanes 0-15, 1 = lanes 16-31
    const int lr   = lane & 15;
    const int wm   = wave >> 1;         // 0..3  (M sub-tile)
    const int wn   = wave & 1;          // 0..1  (N sub-strip of 64)
    const int m0   = blockIdx.y * 64;
    const int n0   = blockIdx.x * 128;

    // loader mapping
    const int arow  = tid >> 3;         // 0..31  (A rows; +32 for second half)
    const int acol4 = (tid & 7) * 4;    // 0..28
    const int brow  = tid >> 5;         // 0..7   (B k-rows; +8,+16,+24)
    const int bcol4 = (tid & 31) * 4;   // 0..124

    v8f acc[4];
    acc[0] = v8f{}; acc[1] = v8f{}; acc[2] = v8f{}; acc[3] = v8f{};

    // ---- prologue: stage chunk 0 into buffer 0 ----
    {
        #pragma unroll
        for (int rr = 0; rr < 2; ++rr) {
            const int row = arow + rr * 32;
            const v4f av = *(const v4f*)(&x[(size_t)(m0 + row) * Dsz + acol4]);
            float* dst = &A_s[0][row * APAD + acol4];
            dst[0] = av.x; dst[1] = av.y; dst[2] = av.z; dst[3] = av.w;
        }
        #pragma unroll
        for (int rr = 0; rr < 4; ++rr) {
            const int row = brow + rr * 8;          // k index 0..31
            const v4f bv = *(const v4f*)(&Wg[(size_t)row * Dsz + n0 + bcol4]);
            float* dst = &B_s[0][(row >> 1) * BPAD + bcol4 * 2 + (row & 1)];
            dst[0] = bv.x; dst[2] = bv.y; dst[4] = bv.z; dst[6] = bv.w;
        }
    }
    __syncthreads();

    for (int kk = 0; kk < Dsz; kk += 32) {
        const int  buf  = (kk >> 5) & 1;
        const bool more = (kk + 32 < Dsz);

        // ---- issue global loads for next chunk (in flight during WMMAs) ----
        v4f anext[2], bnext[4];
        if (more) {
            #pragma unroll
            for (int rr = 0; rr < 2; ++rr) {
                const int row = arow + rr * 32;
                anext[rr] = *(const v4f*)(&x[(size_t)(m0 + row) * Dsz + kk + 32 + acol4]);
            }
            #pragma unroll
            for (int rr = 0; rr < 4; ++rr) {
                const int row = brow + rr * 8;
                bnext[rr] = *(const v4f*)(&Wg[(size_t)(kk + 32 + row) * Dsz + n0 + bcol4]);
            }
        }

        // ---- 8 WMMA K-steps of 4 from current buffer ----
        #pragma unroll
        for (int ks = 0; ks < 8; ++ks) {
            const int k4 = ks * 4 + 2 * half;   // this half-wave's K pair
            const int kp = k4 >> 1;             // kpair row in B layout
            const v2f a = *(const v2f*)(&A_s[buf][(wm * 16 + lr) * APAD + k4]);
            #pragma unroll
            for (int j = 0; j < 4; ++j) {
                const int nb = wn * 64 + j * 16 + lr;
                const v2f b = *(const v2f*)(&B_s[buf][kp * BPAD + nb * 2]);
                acc[j] = __builtin_amdgcn_wmma_f32_16x16x4_f32(
                    false, a, false, b, (short)0, acc[j], false, false);
            }
        }

        // ---- store staged regs into the other buffer ----
        if (more) {
            #pragma unroll
            for (int rr = 0; rr < 2; ++rr) {
                const int row = arow + rr * 32;
                float* dst = &A_s[buf ^ 1][row * APAD + acol4];
                dst[0] = anext[rr].x; dst[1] = anext[rr].y;
                dst[2] = anext[rr].z; dst[3] = anext[rr].w;
            }
            #pragma unroll
            for (int rr = 0; rr < 4; ++rr) {
                const int row = brow + rr * 8;
                float* dst = &B_s[buf ^ 1][(row >> 1) * BPAD + bcol4 * 2 + (row & 1)];
                dst[0] = bnext[rr].x; dst[2] = bnext[rr].y;
                dst[4] = bnext[rr].z; dst[6] = bnext[rr].w;
            }
        }
        __syncthreads();
    }

    // ---- fused epilogue: gated = x * sigmoid(Y + bg) ----
    #pragma unroll
    for (int j = 0; j < 4; ++j) {
        const int n    = n0 + wn * 64 + j * 16 + lr;
        const float bn = bg[n];
        #pragma unroll
        for (int r = 0; r < 8; ++r) {
            const int m  = m0 + wm * 16 + half * 8 + r;  // C/D layout: VGPR r -> M=r (+8 hi half)
            const float yv = acc[j][r] + bn;
            const float g  = 1.0f / (1.0f + expf(-yv));
            const size_t idx = (size_t)m * Dsz + n;
            gated[idx] = x[idx] * g;
        }
    }
}

// ---------------------------------------------------------------------------
// Kernel 2: pooled[b,d] = mean over s of gated[b,s,d]
// grid (1, B), block 256; each thread owns 4 d-columns -> b128 loads.
// ---------------------------------------------------------------------------
__global__ __launch_bounds__(256) void pool_kernel(
    const float* __restrict__ gated, float* __restrict__ pooled)
{
    const int b  = blockIdx.y;
    const int d4 = threadIdx.x * 4;
    v4f s = v4f{};
    for (int t = 0; t < Ssz; ++t) {
        const v4f g = *(const v4f*)(&gated[((size_t)b * Ssz + t) * Dsz + d4]);
        s.x += g.x; s.y += g.y; s.z += g.z; s.w += g.w;
    }
    const float inv = 1.0f / (float)Ssz;
    v4f o; o.x = s.x * inv; o.y = s.y * inv; o.z = s.z * inv; o.w = s.w * inv;
    *(v4f*)(&pooled[b * Dsz + d4]) = o;
}

// ---------------------------------------------------------------------------
// Kernel 3: router MLP + softmax + top-3 renorm -> probs (to d_out) and gains
// Single block, 256 threads. pooled staged in LDS (64KB), h in LDS (16KB).
// ---------------------------------------------------------------------------
__global__ __launch_bounds__(256) void router_kernel(
    const float* __restrict__ pooled,
    const float* __restrict__ W1, const float* __restrict__ b1,
    const float* __restrict__ W2, const float* __restrict__ b2,
    float* __restrict__ probs, float* __restrict__ gains)
{
    __shared__ float pooled_s[Bsz * Dsz];
    __shared__ float h_s[Bsz * Hsz];
    __shared__ float logits_s[Bsz * Esz];

    const int t = threadIdx.x;
    for (int i = t; i < Bsz * Dsz; i += 256) pooled_s[i] = pooled[i];
    __syncthreads();

    // h[:, t] = tanh(pooled @ W1[:, t] + b1[t])
    float acc[Bsz];
    #pragma unroll
    for (int b = 0; b < Bsz; ++b) acc[b] = 0.0f;
    for (int k = 0; k < Dsz; ++k) {
        const float w = W1[(size_t)k * Hsz + t];
        #pragma unroll
        for (int b = 0; b < Bsz; ++b) acc[b] += pooled_s[b * Dsz + k] * w;
    }
    const float bt = b1[t];
    #pragma unroll
    for (int b = 0; b < Bsz; ++b) h_s[b * Hsz + t] = tanhf(acc[b] + bt);
    __syncthreads();

    // logits[b, e]
    if (t < Bsz * Esz) {
        const int b = t >> 3, e = t & 7;
        float a = b2[e];
        for (int j = 0; j < Hsz; ++j) a += h_s[b * Hsz + j] * W2[j * Esz + e];
        logits_s[t] = a;
    }
    __syncthreads();

    // per-batch softmax + top-3 renorm -> gains
    if (t < Bsz) {
        const int b = t;
        float mx = logits_s[b * Esz + 0];
        #pragma unroll
        for (int e = 1; e < Esz; ++e) mx = fmaxf(mx, logits_s[b * Esz + e]);
        float p[Esz]; float sum = 0.0f;
        #pragma unroll
        for (int e = 0; e < Esz; ++e) { p[e] = expf(logits_s[b * Esz + e] - mx); sum += p[e]; }
        const float inv = 1.0f / sum;
        #pragma unroll
        for (int e = 0; e < Esz; ++e) { p[e] *= inv; probs[b * Esz + e] = p[e]; }

        // top-3 (ties -> lowest index, like lax.top_k)
        bool used[Esz] = {false, false, false, false, false, false, false, false};
        int   idx[Ktop]; float w[Ktop]; float wsum = 0.0f;
        #pragma unroll
        for (int k = 0; k < Ktop; ++k) {
            float best = -1.0f; int bi = 0;
            #pragma unroll
            for (int e = 0; e < Esz; ++e)
                if (!used[e] && p[e] > best) { best = p[e]; bi = e; }
            used[bi] = true; idx[k] = bi; w[k] = best; wsum += best;
        }
        float g[Esz] = {0, 0, 0, 0, 0, 0, 0, 0};
        const float winv = 1.0f / wsum;
        #pragma unroll
        for (int k = 0; k < Ktop; ++k) g[idx[k]] += w[k] * winv;
        #pragma unroll
        for (int e = 0; e < Esz; ++e) gains[b * Esz + e] = g[e];
    }
}

// ---------------------------------------------------------------------------
// Kernel 4: routed[e,b,s,d] = gains[b,e] * gated[b,s,d]
// One NT v4f load of gated per thread, 8 non-temporal v4f stores (streaming).
// ---------------------------------------------------------------------------
__global__ __launch_bounds__(256) void broadcast_kernel(
    const float* __restrict__ gated, const float* __restrict__ gains,
    float* __restrict__ routed)
{
    const size_t i4  = (size_t)blockIdx.x * 256 + threadIdx.x;  // float4 index
    const size_t idx = i4 * 4;
    const int b = (int)(idx / ((size_t)Ssz * Dsz));
    const v4f gv = __builtin_nontemporal_load((const v4f*)(gated + idx));
    #pragma unroll
    for (int e = 0; e < Esz; ++e) {
        const float gn = gains[b * Esz + e];
        v4f o; o.x = gv.x * gn; o.y = gv.y * gn; o.z = gv.z * gn; o.w = gv.w * gn;
        v4f* dst = (v4f*)(routed + (size_t)e * Bsz * Ssz * Dsz) + i4;
        __builtin_nontemporal_store(o, dst);
    }
}

// ---------------------------------------------------------------------------
extern "C" void kernel_launch(void* const* d_in, const int* in_sizes, int n_in,
                              void* d_out, int out_size, void* d_ws, size_t ws_size,
                              hipStream_t stream)
{
    (void)in_sizes; (void)n_in; (void)out_size; (void)ws_size;
    const float* x  = (const float*)d_in[0];
    const float* Wg = (const float*)d_in[1];
    const float* bg = (const float*)d_in[2];
    const float* W1 = (const float*)d_in[3];
    const float* b1 = (const float*)d_in[4];
    const float* W2 = (const float*)d_in[5];
    const float* b2 = (const float*)d_in[6];

    float* routed = (float*)d_out;                              // [E,B,S,D]
    float* probs  = routed + (size_t)Esz * Bsz * Ssz * Dsz;     // [B,E]

    float* gated  = (float*)d_ws;                               // [B,S,D] 64MB
    float* pooled = gated + (size_t)Msz * Dsz;                  // [B,D]
    float* gains  = pooled + Bsz * Dsz;                         // [B,E]

    gate_gemm_kernel<<<dim3(Dsz / 128, Msz / 64), 256, 0, stream>>>(x, Wg, bg, gated);
    pool_kernel<<<dim3(1, Bsz), 256, 0, stream>>>(gated, pooled);
    router_kernel<<<1, 256, 0, stream>>>(pooled, W1, b1, W2, b2, probs, gains);
    broadcast_kernel<<<(Bsz * Ssz * Dsz / 4) / 256, 256, 0, stream>>>(gated, gains, routed);
}